// VVCUpsampler_14516989460645
// MI455X (gfx1250) — compile-verified
//
#include <hip/hip_runtime.h>
#include <hip/hip_bf16.h>

typedef float v2f __attribute__((ext_vector_type(2)));
typedef float v8f __attribute__((ext_vector_type(8)));

#define LSTR 20   // LDS row stride in floats: conflict-free & 8B-aligned reads

// lane <-> lane^16 exchange: DS_SWIZZLE_B32 SWAPX16 (xor=0x10, and=0x1f -> 0x401F)
__device__ __forceinline__ float half_swap(float v) {
#if __has_builtin(__builtin_amdgcn_ds_swizzle)
    return __builtin_bit_cast(float,
        __builtin_amdgcn_ds_swizzle(__builtin_bit_cast(int, v), 0x401F));
#else
    return __shfl_xor(v, 16, 32);
#endif
}

// One wave32 = one 16x16 output tile = M * In * M^T  via 8x V_WMMA_F32_16X16X4_F32.
__global__ __launch_bounds__(256) void VVCUpsampler_wmma_kernel(
    const float* __restrict__ src0,   // (planes, 512, 512)
    const float* __restrict__ wh,     // (2, 8) horizontal polyphase taps (already /64)
    const float* __restrict__ wv,     // (2, 8) vertical polyphase taps
    float* __restrict__ out,          // (planes, 1024, 1024)
    int nplanes)
{
    __shared__ float lds[8][16 * LSTR];

    const int lane = threadIdx.x & 31;
    const int wave = threadIdx.x >> 5;
    const int q    = lane & 15;   // row (A role) / col (B,D role)
    const int hi   = lane >> 4;   // half-wave: splits K (A/B) or M (D)
    (void)nplanes;

    const long tile = (long)blockIdx.x * 8 + wave;
    const int tx  = (int)(tile & 63);
    const int ty  = (int)((tile >> 6) & 63);
    const int img = (int)(tile >> 12);

    const int x0 = tx * 8;
    const int y0 = ty * 8;
    const float* __restrict__ src = src0 + (size_t)img * (512 * 512);
    float* myl = lds[wave];

    // ---- Stage 16x16 edge-clamped input patch (top-left at (y0-3, x0-3)) into
    // this wave's private LDS region using the CDNA5 async direct-to-LDS path:
    // GLOBAL_LOAD_ASYNC_TO_LDS_B32 (no VGPR round-trip, no ds_store, ASYNCcnt).
    // Each half-wave covers one contiguous 64B row segment per issue (coalesced).
    {
        int gx = x0 - 3 + q;
        gx = gx < 0 ? 0 : (gx > 511 ? 511 : gx);
#pragma unroll
        for (int r = 0; r < 8; ++r) {
            const int i = 2 * r + hi;              // patch row 0..15
            int gy = y0 - 3 + i;
            gy = gy < 0 ? 0 : (gy > 511 ? 511 : gy);
            const float* gp = src + (size_t)gy * 512 + gx;
            // low 32 bits of a flat shared pointer == LDS offset (aperture layout)
            unsigned ldsoff = (unsigned)(size_t)&myl[i * LSTR + q];
            unsigned long long gaddr = (unsigned long long)(size_t)gp;
            asm volatile("global_load_async_to_lds_b32 %0, %1, off"
                         :: "v"(ldsoff), "v"(gaddr)
                         : "memory");
        }
    }

    // ---- Per-lane strip of the banded polyphase matrix M[r][k] = coeff[r&1][k-(r>>1)].
    // Same per-lane values serve as B-operand (GEMM1, lane=q is N) and A-operand
    // (GEMM2, lane=q is M) because both layouts put lane=q, halves split K.
    float mh[8], mvv[8];
#pragma unroll
    for (int c = 0; c < 4; ++c) {
#pragma unroll
        for (int d = 0; d < 2; ++d) {
            const int k   = 4 * c + 2 * hi + d;
            const int tap = k - (q >> 1);
            float fh = 0.0f, fv = 0.0f;
            if (tap >= 0 && tap < 8) {
                fh = wh[(q & 1) * 8 + tap];
                fv = wv[(q & 1) * 8 + tap];
            }
            mh[2 * c + d]  = fh;
            mvv[2 * c + d] = fv;
        }
    }

    // Wave consumes only its own LDS region: a wave-local ASYNCcnt wait is the
    // only ordering needed (no block barrier, waves don't convoy).
    asm volatile("s_wait_asynccnt 0x0" ::: "memory");

    // ---- GEMM1: T = In * Mh^T   (A = patch rows from LDS, B = mh strip)
    v8f t = {0.f, 0.f, 0.f, 0.f, 0.f, 0.f, 0.f, 0.f};
#pragma unroll
    for (int c = 0; c < 4; ++c) {
        const int k = 4 * c + 2 * hi;
        v2f a = { myl[q * LSTR + k], myl[q * LSTR + k + 1] };
        v2f b = { mh[2 * c], mh[2 * c + 1] };
        t = __builtin_amdgcn_wmma_f32_16x16x4_f32(false, a, false, b,
                                                  (short)0, t, false, false);
    }

    // ---- Re-stripe T from D layout (halves split M) to B layout (halves split K):
    // single-op half-wave swap via DS_SWIZZLE SWAPX16.
    float dv[8], xv[8];
#pragma unroll
    for (int j = 0; j < 8; ++j) dv[j] = t[j];
#pragma unroll
    for (int j = 0; j < 8; ++j) xv[j] = half_swap(dv[j]);
    float b2[8];
    b2[0] = hi ? xv[2] : dv[0];
    b2[1] = hi ? xv[3] : dv[1];
    b2[2] = hi ? xv[6] : dv[4];
    b2[3] = hi ? xv[7] : dv[5];
    b2[4] = hi ? dv[2] : xv[0];
    b2[5] = hi ? dv[3] : xv[1];
    b2[6] = hi ? dv[6] : xv[4];
    b2[7] = hi ? dv[7] : xv[5];

    // ---- GEMM2: Out = Mv * T
    v8f o = {0.f, 0.f, 0.f, 0.f, 0.f, 0.f, 0.f, 0.f};
#pragma unroll
    for (int c = 0; c < 4; ++c) {
        v2f a = { mvv[2 * c], mvv[2 * c + 1] };
        v2f b = { b2[2 * c], b2[2 * c + 1] };
        o = __builtin_amdgcn_wmma_f32_16x16x4_f32(false, a, false, b,
                                                  (short)0, o, false, false);
    }

    // ---- Store 16x16 output tile. Output (201MB) is streaming & never re-read:
    // non-temporal stores keep it from evicting the reusable input halo in L2.
    float* __restrict__ dst = out + (size_t)img * (1024 * 1024)
                                  + (size_t)(2 * y0) * 1024 + (size_t)(2 * x0);
#pragma unroll
    for (int j = 0; j < 8; ++j) {
        __builtin_nontemporal_store(o[j], &dst[(size_t)(8 * hi + j) * 1024 + q]);
    }
}

extern "C" void kernel_launch(void* const* d_in, const int* in_sizes, int n_in,
                              void* d_out, int out_size, void* d_ws, size_t ws_size,
                              hipStream_t stream) {
    const float* x  = (const float*)d_in[0];   // (16,3,512,512) fp32
    const float* wh = (const float*)d_in[1];   // (2,1,1,8) fp32
    const float* wv = (const float*)d_in[2];   // (2,1,8,1) fp32
    float* out = (float*)d_out;                // (16,3,1024,1024) fp32

    const int planes = in_sizes[0] / (512 * 512);   // 48
    const int tiles  = planes * 64 * 64;            // one 16x16 out tile per wave
    const int blocks = tiles / 8;                   // 8 waves (tiles) per 256-thr block

    VVCUpsampler_wmma_kernel<<<blocks, 256, 0, stream>>>(x, wh, wv, out, planes);
}